// TopKRouter_15745350107278
// MI455X (gfx1250) — compile-verified
//
#include <hip/hip_runtime.h>

#define D_MODEL   2048
#define N_EXPERTS 64
#define TOP_K     8
#define N_TOKENS  16384

// Output layout (flat float, in reference return order)
#define OFF_IDX   0
#define OFF_WTS   (N_TOKENS * TOP_K)                 // 131072
#define OFF_PROBS (2 * N_TOKENS * TOP_K)             // 262144
#define OFF_LOGIT (OFF_PROBS + N_TOKENS * N_EXPERTS) // 1310720

typedef float v2f __attribute__((ext_vector_type(2)));
typedef float v8f __attribute__((ext_vector_type(8)));

#define CK          32          // K-chunk per LDS buffer
#define AB_STRIDE   36          // 36*e mod 64 covers multiples of 4 -> conflict-free b64 reads
#define AB_BUF      (64 * AB_STRIDE)   // 2304 floats per buffer
#define LDSA_OFF    0                  // A0 | A1
#define LDSB_OFF    (2 * AB_BUF)       // B0 | B1
#define LDSL_OFF    (4 * AB_BUF)       // logits 64 x 66
#define LDSL_STRIDE 66
#define LDS_FLOATS  (4 * AB_BUF + 64 * LDSL_STRIDE)  // 13440 floats = 53760 B

#define N_CHUNKS (D_MODEL / CK)  // 64

// Async global -> LDS DMA (gfx1250), tracked by ASYNCcnt; no VGPR transit.
__device__ __forceinline__ void async_load_b128(unsigned int lds_byte_off,
                                                const float* gaddr) {
    asm volatile("global_load_async_to_lds_b128 %0, %1, off"
                 :: "v"(lds_byte_off), "v"(gaddr)
                 : "memory");
}

__device__ __forceinline__ void wait_async0() {
    asm volatile("s_wait_asynccnt 0x0" ::: "memory");
}

__global__ __launch_bounds__(256)
void TopKRouter_gemm_topk_kernel(const float* __restrict__ x,
                                 const float* __restrict__ gw,
                                 float* __restrict__ out)
{
    extern __shared__ float lds[];
    float* ldsL = lds + LDSL_OFF;

    const int tid  = (int)threadIdx.x;
    const int lane = tid & 31;
    const int w    = tid >> 5;      // wave 0..7
    const int g    = w >> 1;        // token group 0..3 (16 tokens each)
    const int h    = w & 1;         // expert half 0..1 (2 N-tiles each)

    const int blockTok0 = (int)blockIdx.x * 64;
    const int tok0      = blockTok0 + g * 16;

    const int lrow  = lane & 15;         // M (A) / N (B) index
    const int khalf = (lane >> 4) << 1;  // lanes 0-15 -> K{0,1}; 16-31 -> K{2,3}

    // Fragment read bases (float index into lds[]), buffer 0
    const int aIdx  = LDSA_OFF + (g * 16 + lrow) * AB_STRIDE + khalf;
    const int bIdx0 = LDSB_OFF + ((2 * h + 0) * 16 + lrow) * AB_STRIDE + khalf;
    const int bIdx1 = LDSB_OFF + ((2 * h + 1) * 16 + lrow) * AB_STRIDE + khalf;

    // Per-thread async staging assignment: 2 rows-of-16B per tile per thread
    // idx = i*256 + tid : row = idx>>3 (0..63), kq = (idx&7)*4 (0..28)
    const float* xRow0  = x  + (size_t)(blockTok0 + (tid >> 3)) * D_MODEL + ((tid & 7) << 2);
    const float* xRow1  = x  + (size_t)(blockTok0 + ((256 + tid) >> 3)) * D_MODEL + ((tid & 7) << 2);
    const float* gwRow0 = gw + (size_t)(tid >> 3) * D_MODEL + ((tid & 7) << 2);
    const float* gwRow1 = gw + (size_t)((256 + tid) >> 3) * D_MODEL + ((tid & 7) << 2);
    const unsigned int ldsStage0 =
        (unsigned int)(((tid >> 3) * AB_STRIDE + ((tid & 7) << 2)) * 4);
    const unsigned int ldsStage1 =
        (unsigned int)(((((256 + tid) >> 3)) * AB_STRIDE + ((tid & 7) << 2)) * 4);

    v8f acc0 = {};
    v8f acc1 = {};

    // ---- issue chunk 0 ----
    {
        const unsigned int bufA = LDSA_OFF * 4u;
        const unsigned int bufB = LDSB_OFF * 4u;
        async_load_b128(bufA + ldsStage0, xRow0);
        async_load_b128(bufA + ldsStage1, xRow1);
        async_load_b128(bufB + ldsStage0, gwRow0);
        async_load_b128(bufB + ldsStage1, gwRow1);
    }

    for (int c = 0; c < N_CHUNKS; ++c) {
        wait_async0();        // this wave's DMA for chunk c complete
        __syncthreads();      // everyone's DMA complete; prev compute done

        if (c + 1 < N_CHUNKS) {
            const int kb = (c + 1) * CK;
            const unsigned int bufA = (unsigned int)((LDSA_OFF + ((c + 1) & 1) * AB_BUF) * 4);
            const unsigned int bufB = (unsigned int)((LDSB_OFF + ((c + 1) & 1) * AB_BUF) * 4);
            async_load_b128(bufA + ldsStage0, xRow0 + kb);
            async_load_b128(bufA + ldsStage1, xRow1 + kb);
            async_load_b128(bufB + ldsStage0, gwRow0 + kb);
            async_load_b128(bufB + ldsStage1, gwRow1 + kb);
        }

        const int bo = (c & 1) * AB_BUF;
        const float* aB  = lds + aIdx  + bo;
        const float* bB0 = lds + bIdx0 + bo;
        const float* bB1 = lds + bIdx1 + bo;

        #pragma unroll
        for (int kk = 0; kk < CK; kk += 4) {
            v2f a  = *(const v2f*)(aB  + kk);
            v2f b0 = *(const v2f*)(bB0 + kk);
            v2f b1 = *(const v2f*)(bB1 + kk);
            acc0 = __builtin_amdgcn_wmma_f32_16x16x4_f32(false, a, false, b0,
                                                         (short)0, acc0, false, false);
            acc1 = __builtin_amdgcn_wmma_f32_16x16x4_f32(false, a, false, b1,
                                                         (short)0, acc1, false, false);
        }
    }

    // Epilogue: C layout -> VGPR r holds M = r + 8*(lane>=16), N = lane%16
    float* logits_out = out + OFF_LOGIT;
    const int rowAdd = (lane >> 4) << 3;
    const int e0 = (2 * h + 0) * 16 + lrow;
    const int e1 = (2 * h + 1) * 16 + lrow;
    #pragma unroll
    for (int r = 0; r < 8; ++r) {
        int row = r + rowAdd;                 // 0..15 within token group
        int tok = tok0 + row;
        float v0 = acc0[r];
        float v1 = acc1[r];
        ldsL[(g * 16 + row) * LDSL_STRIDE + e0] = v0;
        ldsL[(g * 16 + row) * LDSL_STRIDE + e1] = v1;
        logits_out[(size_t)tok * N_EXPERTS + e0] = v0;
        logits_out[(size_t)tok * N_EXPERTS + e1] = v1;
    }
    __syncthreads();

    // One thread per token: top-8 + softmaxes
    if (tid < 64) {
        float* rowL = ldsL + tid * LDSL_STRIDE;
        const int tok = blockTok0 + tid;

        float bval[TOP_K];
        int   bidx[TOP_K];
        #pragma unroll
        for (int j = 0; j < TOP_K; ++j) { bval[j] = -3.402823466e38f; bidx[j] = 0; }

        for (int e = 0; e < N_EXPERTS; ++e) {
            float v  = rowL[e];
            int   ei = e;
            #pragma unroll
            for (int j = 0; j < TOP_K; ++j) {
                bool  gt = v > bval[j];
                float nv = gt ? bval[j] : v;
                int   ni = gt ? bidx[j] : ei;
                bval[j]  = gt ? v  : bval[j];
                bidx[j]  = gt ? ei : bidx[j];
                v = nv; ei = ni;
            }
        }

        const float LOG2E = 1.44269504088896340736f;
        const float mx = bval[0];   // global max is in the top-8

        // Full softmax: overwrite LDS row with exps, then normalize
        float sum = 0.f;
        for (int e = 0; e < N_EXPERTS; ++e) {
            float p = __builtin_amdgcn_exp2f((rowL[e] - mx) * LOG2E);
            rowL[e] = p;
            sum += p;
        }
        float rinv = __builtin_amdgcn_rcpf(sum);
        float* probs_out = out + OFF_PROBS + (size_t)tok * N_EXPERTS;
        for (int e = 0; e < N_EXPERTS; ++e) probs_out[e] = rowL[e] * rinv;

        // Top-k softmax over the 8 selected logits
        float es[TOP_K];
        float s8 = 0.f;
        #pragma unroll
        for (int j = 0; j < TOP_K; ++j) {
            es[j] = __builtin_amdgcn_exp2f((bval[j] - mx) * LOG2E);
            s8 += es[j];
        }
        float r8 = __builtin_amdgcn_rcpf(s8);
        float* idx_out = out + OFF_IDX + (size_t)tok * TOP_K;
        float* wt_out  = out + OFF_WTS + (size_t)tok * TOP_K;
        #pragma unroll
        for (int j = 0; j < TOP_K; ++j) {
            idx_out[j] = (float)bidx[j];
            wt_out[j]  = es[j] * r8;
        }
    }
}

extern "C" void kernel_launch(void* const* d_in, const int* in_sizes, int n_in,
                              void* d_out, int out_size, void* d_ws, size_t ws_size,
                              hipStream_t stream) {
    (void)in_sizes; (void)n_in; (void)out_size; (void)d_ws; (void)ws_size;
    const float* x  = (const float*)d_in[0];   // [16384, 2048] f32
    const float* gw = (const float*)d_in[1];   // [64, 2048] f32
    float* out = (float*)d_out;

    dim3 grid(N_TOKENS / 64);   // 256 blocks
    dim3 block(256);            // 8 waves (wave32)
    size_t smem = LDS_FLOATS * sizeof(float);  // 53760 B
    TopKRouter_gemm_topk_kernel<<<grid, block, smem, stream>>>(x, gw, out);
}